// GNN_70884140253871
// MI455X (gfx1250) — compile-verified
//
#include <hip/hip_runtime.h>

#define GRAPH_NUM 3
#define N_USER 100000
#define N_ITEM 50000
#define LATDIM 128
#define N_EDGES 1000000
#define GNN_LAYERS 2
#define EPB 512            // edges staged per LDS chunk
#define ROWS_PER_BLOCK 16  // output rows owned by one block

// gfx1250 async global->LDS path (ASYNCcnt-tracked), guarded so host pass and
// older toolchains fall back to plain loads.
#if defined(__has_builtin)
#  if __has_builtin(__builtin_amdgcn_global_load_async_to_lds_b32)
#    define HAVE_ASYNC_LDS 1
#  endif
#endif

// The builtin on this toolchain takes typed (int) pointers:
//   (addrspace(1) int* gaddr, addrspace(3) int* lds, imm offset, imm cpol)
typedef __attribute__((address_space(1))) int GlobalInt;
typedef __attribute__((address_space(3))) int LdsInt;

__device__ __forceinline__ void stage_indices(const int* __restrict__ gsrc,
                                              const int* __restrict__ gtgt,
                                              int* __restrict__ ssrc,
                                              int* __restrict__ stgt,
                                              int m, int t)
{
#ifdef HAVE_ASYNC_LDS
    for (int i = t; i < m; i += LATDIM) {
        __builtin_amdgcn_global_load_async_to_lds_b32(
            (GlobalInt*)(gsrc + i), (LdsInt*)(ssrc + i), 0, 0);
        __builtin_amdgcn_global_load_async_to_lds_b32(
            (GlobalInt*)(gtgt + i), (LdsInt*)(stgt + i), 0, 0);
    }
#  if __has_builtin(__builtin_amdgcn_s_wait_asynccnt)
    __builtin_amdgcn_s_wait_asynccnt(0);
#  else
    asm volatile("s_wait_asynccnt 0" ::: "memory");
#  endif
#else
    for (int i = t; i < m; i += LATDIM) {
        ssrc[i] = gsrc[i];
        stgt[i] = gtgt[i];
    }
#endif
}

// One block owns output rows [r0, r1). tgt_idx is sorted, so the block's edges
// form one contiguous range found by binary search. One thread per feature dim
// (blockDim.x == 128): row reads are fully coalesced 512B transactions, the
// per-row accumulator lives in a register, and each msg row is written exactly
// once by exactly one block -> deterministic, no atomics, no pre-zeroing.
__global__ __launch_bounds__(LATDIM) void seg_sum_kernel(
    const float* __restrict__ src_emb,   // [n_src, 128]
    const int*   __restrict__ src_idx,   // [n_edges]
    const int*   __restrict__ tgt_idx,   // [n_edges], sorted ascending
    float*       __restrict__ msg,       // [n_rows, 128]
    int n_edges, int n_rows)
{
    const int t  = threadIdx.x;
    const int r0 = blockIdx.x * ROWS_PER_BLOCK;
    int r1 = r0 + ROWS_PER_BLOCK;
    if (r1 > n_rows) r1 = n_rows;

    // lower_bound(tgt_idx, r0); uniform across the block (broadcast loads)
    int lo = 0, hi = n_edges;
    while (lo < hi) {
        int mid = (lo + hi) >> 1;
        if (tgt_idx[mid] < r0) lo = mid + 1; else hi = mid;
    }
    int e = lo;

    __shared__ int s_src[EPB];
    __shared__ int s_tgt[EPB];

    int   row = r0;
    float acc = 0.0f;

    for (;;) {
        int rem = n_edges - e;
        int m = rem < EPB ? rem : EPB;
        if (m <= 0) break;

        stage_indices(src_idx + e, tgt_idx + e, s_src, s_tgt, m, t);
        __syncthreads();

        bool hit_end = false;
        for (int i = 0; i < m; ++i) {
            int tg = s_tgt[i];                 // LDS broadcast, uniform branch
            if (tg >= r1) { hit_end = true; break; }
            if (tg != row) {
                msg[row * LATDIM + t] = acc;
                acc = 0.0f;
                for (int r = row + 1; r < tg; ++r) msg[r * LATDIM + t] = 0.0f;
                row = tg;
            }
            acc += src_emb[s_src[i] * LATDIM + t];
        }
        __syncthreads();                       // protect LDS before restage
        if (hit_end) break;
        e += m;
    }

    // flush current row and any trailing empty rows in [row+1, r1)
    msg[row * LATDIM + t] = acc;
    for (int r = row + 1; r < r1; ++r) msg[r * LATDIM + t] = 0.0f;
}

// cur = cur + leaky(msg); sum += cur      (b128 vectorized, in place)
__global__ __launch_bounds__(256) void update_kernel(
    const float4* __restrict__ msg,
    float4*       __restrict__ cur,
    float4*       __restrict__ sum,
    int n4)
{
    int i = blockIdx.x * 256 + threadIdx.x;
    if (i >= n4) return;
    float4 m = msg[i];
    float4 c = cur[i];
    float4 s = sum[i];
    c.x += fmaxf(m.x, 0.01f * m.x);
    c.y += fmaxf(m.y, 0.01f * m.y);
    c.z += fmaxf(m.z, 0.01f * m.z);
    c.w += fmaxf(m.w, 0.01f * m.w);
    s.x += c.x; s.y += c.y; s.z += c.z; s.w += c.w;
    cur[i] = c;
    sum[i] = s;
}

extern "C" void kernel_launch(void* const* d_in, const int* in_sizes, int n_in,
                              void* d_out, int out_size, void* d_ws, size_t ws_size,
                              hipStream_t stream)
{
    const float* ue      = (const float*)d_in[0];  // [3, 100000, 128]
    const float* ie      = (const float*)d_in[1];  // [3,  50000, 128]
    const int*   adj_src = (const int*)  d_in[2];  // [3, 1e6] item indices
    const int*   adj_tgt = (const int*)  d_in[3];  // [3, 1e6] user indices (sorted)
    const int*   tp_src  = (const int*)  d_in[4];  // [3, 1e6] user indices
    const int*   tp_tgt  = (const int*)  d_in[5];  // [3, 1e6] item indices (sorted)

    const size_t uElems = (size_t)N_USER * LATDIM;   // 12.8M floats
    const size_t iElems = (size_t)N_ITEM * LATDIM;   //  6.4M floats

    float* out_user = (float*)d_out;                           // [3,U,128]
    float* out_item = out_user + (size_t)GRAPH_NUM * uElems;   // [3,I,128]

    // workspace: u_cur | i_cur | u_msg | i_msg  (~154 MB), reused per graph
    const size_t need = 2 * (uElems + iElems) * sizeof(float);
    if (ws_size < need) return;
    float* u_cur = (float*)d_ws;
    float* i_cur = u_cur + uElems;
    float* u_msg = i_cur + iElems;
    float* i_msg = u_msg + uElems;

    // sum accumulators start at the initial embeddings
    (void)hipMemcpyAsync(out_user, ue, (size_t)GRAPH_NUM * uElems * sizeof(float),
                         hipMemcpyDeviceToDevice, stream);
    (void)hipMemcpyAsync(out_item, ie, (size_t)GRAPH_NUM * iElems * sizeof(float),
                         hipMemcpyDeviceToDevice, stream);

    const int uBlocks = (N_USER + ROWS_PER_BLOCK - 1) / ROWS_PER_BLOCK;  // 6250
    const int iBlocks = (N_ITEM + ROWS_PER_BLOCK - 1) / ROWS_PER_BLOCK;  // 3125
    const int un4 = (int)(uElems / 4);
    const int in4 = (int)(iElems / 4);

    for (int g = 0; g < GRAPH_NUM; ++g) {
        (void)hipMemcpyAsync(u_cur, ue + (size_t)g * uElems, uElems * sizeof(float),
                             hipMemcpyDeviceToDevice, stream);
        (void)hipMemcpyAsync(i_cur, ie + (size_t)g * iElems, iElems * sizeof(float),
                             hipMemcpyDeviceToDevice, stream);

        const int* aS = adj_src + (size_t)g * N_EDGES;
        const int* aT = adj_tgt + (size_t)g * N_EDGES;
        const int* tS = tp_src  + (size_t)g * N_EDGES;
        const int* tT = tp_tgt  + (size_t)g * N_EDGES;
        float* uSum = out_user + (size_t)g * uElems;
        float* iSum = out_item + (size_t)g * iElems;

        for (int L = 0; L < GNN_LAYERS; ++L) {
            // both message passes read the *old* cur buffers
            seg_sum_kernel<<<uBlocks, LATDIM, 0, stream>>>(i_cur, aS, aT, u_msg,
                                                           N_EDGES, N_USER);
            seg_sum_kernel<<<iBlocks, LATDIM, 0, stream>>>(u_cur, tS, tT, i_msg,
                                                           N_EDGES, N_ITEM);
            // then apply updates in place
            update_kernel<<<(un4 + 255) / 256, 256, 0, stream>>>(
                (const float4*)u_msg, (float4*)u_cur, (float4*)uSum, un4);
            update_kernel<<<(in4 + 255) / 256, 256, 0, stream>>>(
                (const float4*)i_msg, (float4*)i_cur, (float4*)iSum, in4);
        }
    }
}